// TopKGatingNetwork_68813966016589
// MI455X (gfx1250) — compile-verified
//
#include <hip/hip_runtime.h>
#include <hip/hip_bf16.h>
#include <math.h>

typedef __attribute__((ext_vector_type(2))) float v2f;
typedef __attribute__((ext_vector_type(8))) float v8f;

#define INPUT_DIM 2048
#define NUM_EXPERTS 64
#define TOKENS (4 * 4096)
#define CHUNK_K 64
#define LDS_STRIDE 66   // pad to kill bank conflicts on ds_load_b64

// better(a, b): strict "a ranks before b" with JAX top_k tie-break (lower index wins)
__device__ __forceinline__ bool better(float va, int ia, float vb, int ib) {
    return (va > vb) || (va == vb && ia < ib);
}

__global__ __launch_bounds__(256)
void topk_gating_kernel(const float* __restrict__ x,
                        const float* __restrict__ W,
                        const float* __restrict__ bias,
                        float* __restrict__ out) {
    __shared__ float sW[NUM_EXPERTS * LDS_STRIDE];  // W chunk, stored [col][k], ~16.5 KB

    const int tid  = threadIdx.x;
    const int lane = tid & 31;
    const int wave = tid >> 5;
    const int c    = lane & 15;          // column-within-tile / row-within-tile
    const int off  = (lane >> 4) * 2;    // K sub-offset for upper half-wave (A/B layout: V0=K0|K2, V1=K1|K3)

    const int rowBase = blockIdx.x * 128 + wave * 16;  // 8 waves x 16 tokens
    const float* xrow = x + (size_t)(rowBase + c) * INPUT_DIM;

    v8f acc[4] = {};  // 16x16 f32 D tiles for expert columns [0:16),[16:32),[32:48),[48:64)

    for (int kc0 = 0; kc0 < INPUT_DIM; kc0 += CHUNK_K) {
        // ---- stage 64x64 f32 chunk of W into LDS, transposed to [col][k] ----
        const float* Wc = W + (size_t)kc0 * NUM_EXPERTS;  // 4096 contiguous floats
        #pragma unroll
        for (int j = 0; j < 4; ++j) {
            int g = j * 1024 + tid * 4;
            float4 wv = *(const float4*)(Wc + g);
            int k = g >> 6;
            int colb = g & 63;
            sW[(colb + 0) * LDS_STRIDE + k] = wv.x;
            sW[(colb + 1) * LDS_STRIDE + k] = wv.y;
            sW[(colb + 2) * LDS_STRIDE + k] = wv.z;
            sW[(colb + 3) * LDS_STRIDE + k] = wv.w;
        }
        __syncthreads();

        // ---- 16 K-steps of 4, each feeding 4 WMMAs (one per expert tile) ----
        #pragma unroll 4
        for (int ks = 0; ks < CHUNK_K; ks += 4) {
            // A fragment: row (rowBase+c), K = kc0+ks+off .. +1   (8B-aligned float2)
            v2f a = *(const v2f*)(xrow + kc0 + ks + off);
            #pragma unroll
            for (int t = 0; t < 4; ++t) {
                int col = t * 16 + c;
                // B fragment: W[ks+off][col], W[ks+off+1][col] -> contiguous in [col][k] layout
                v2f bf = *(const v2f*)(&sW[col * LDS_STRIDE + ks + off]);
                acc[t] = __builtin_amdgcn_wmma_f32_16x16x4_f32(
                    /*neg_a=*/false, a, /*neg_b=*/false, bf,
                    /*c_mod=*/(short)0, acc[t],
                    /*reuse_a=*/false, /*reuse_b=*/false);
            }
        }
        __syncthreads();
    }

    // ---- bias add: every element of acc[t] in this lane is expert column t*16+c ----
    #pragma unroll
    for (int t = 0; t < 4; ++t) {
        float bv = bias[t * 16 + c];
        acc[t] = acc[t] + bv;
    }

    // ---- per-row top-2 + softmax + sparse scatter ----
    // acc[t] element r: lanes 0-15 -> row r, lanes 16-31 -> row r+8 (C/D layout)
    #pragma unroll
    for (int r = 0; r < 8; ++r) {
        // lane-local top-2 over this lane's 4 expert candidates
        float v1 = acc[0][r];       int i1 = c;
        float v2 = -INFINITY;       int i2 = 0x40000000;
        #pragma unroll
        for (int t = 1; t < 4; ++t) {
            float v = acc[t][r];
            int   i = t * 16 + c;
            if (better(v, i, v1, i1)) { v2 = v1; i2 = i1; v1 = v; i1 = i; }
            else if (better(v, i, v2, i2)) { v2 = v; i2 = i; }
        }
        // butterfly reduce across the 16 lanes of this half-wave (masks < 16 keep halves separate)
        #pragma unroll
        for (int m = 1; m < 16; m <<= 1) {
            float ov1 = __shfl_xor(v1, m, 32); int oi1 = __shfl_xor(i1, m, 32);
            float ov2 = __shfl_xor(v2, m, 32); int oi2 = __shfl_xor(i2, m, 32);
            if (better(v1, i1, ov1, oi1)) {
                // our #1 stays; new #2 = best of (our #2, their #1)
                if (better(ov1, oi1, v2, i2)) { v2 = ov1; i2 = oi1; }
            } else {
                // their #1 wins; new #2 = best of (our #1, their #2)
                if (better(v1, i1, ov2, oi2)) { v2 = v1; i2 = i1; }
                else                          { v2 = ov2; i2 = oi2; }
                v1 = ov1; i1 = oi1;
            }
        }
        // softmax over the two selected logits (v1 >= v2)
        float e  = expf(v2 - v1);
        float w1 = 1.0f / (1.0f + e);
        float w2 = e / (1.0f + e);

        int row = rowBase + r + ((lane >> 4) << 3);
        float* orow = out + (size_t)row * NUM_EXPERTS;
        #pragma unroll
        for (int t = 0; t < 4; ++t) {
            int n = t * 16 + c;
            orow[n] = (n == i1) ? w1 : ((n == i2) ? w2 : 0.0f);
        }
    }
}

extern "C" void kernel_launch(void* const* d_in, const int* in_sizes, int n_in,
                              void* d_out, int out_size, void* d_ws, size_t ws_size,
                              hipStream_t stream) {
    const float* x    = (const float*)d_in[0];  // [4,4096,2048]
    const float* W    = (const float*)d_in[1];  // [2048,64]
    const float* bias = (const float*)d_in[2];  // [64]
    float* out = (float*)d_out;                 // [4,4096,64]

    dim3 grid(TOKENS / 128);  // 128 blocks, each covers 128 tokens
    dim3 block(256);          // 8 waves of 32
    topk_gating_kernel<<<grid, block, 0, stream>>>(x, W, bias, out);
}